// ScoreHead_50586124812459
// MI455X (gfx1250) — compile-verified
//
#include <hip/hip_runtime.h>
#include <math.h>

#define Bn 2
#define Cn 32
#define Hn 512
#define Wn 512
#define Nn 65536
#define PAD 5
#define HWn (Hn * Wn)

typedef __attribute__((ext_vector_type(2))) float v2f;
typedef __attribute__((ext_vector_type(8))) float v8f;

// ---------------- helpers ----------------

__device__ __forceinline__ float wave_max32(float v) {
  #pragma unroll
  for (int o = 16; o > 0; o >>= 1) v = fmaxf(v, __shfl_xor(v, o, 32));
  return v;
}

// ---------------- kernel 1: clear ID grid ----------------

__global__ void __launch_bounds__(256) init_grid_kernel(int* __restrict__ grid) {
  int i = blockIdx.x * blockDim.x + threadIdx.x;
  if (i < Bn * HWn) grid[i] = -1;
}

// ---------------- kernel 2: scatter point ids ----------------

__global__ void __launch_bounds__(256) scatter_ids_kernel(const long long* __restrict__ idxs,
                                                          int* __restrict__ grid) {
  int n = blockIdx.x * blockDim.x + threadIdx.x;
  if (n < Nn) {
    int b = (int)idxs[n * 3 + 0];
    int h = (int)idxs[n * 3 + 1];
    int w = (int)idxs[n * 3 + 2];
    grid[b * HWn + h * Wn + w] = n;  // indices unique by construction
  }
}

// ---------------- kernels 3/4: global max of features ----------------

__global__ void __launch_bounds__(256) reduce_max1_kernel(const float* __restrict__ f,
                                                          float* __restrict__ partial,
                                                          int total) {
  __shared__ float sm[8];
  float m = -INFINITY;
  for (int i = blockIdx.x * blockDim.x + threadIdx.x; i < total; i += gridDim.x * blockDim.x)
    m = fmaxf(m, f[i]);
  m = wave_max32(m);
  int wid = threadIdx.x >> 5;
  if ((threadIdx.x & 31) == 0) sm[wid] = m;
  __syncthreads();
  if (wid == 0) {
    float x = (threadIdx.x < (blockDim.x >> 5)) ? sm[threadIdx.x] : -INFINITY;
    x = wave_max32(x);
    if (threadIdx.x == 0) partial[blockIdx.x] = x;
  }
}

__global__ void __launch_bounds__(256) reduce_max2_kernel(float* __restrict__ partial, int nblk) {
  __shared__ float sm[8];
  float m = -INFINITY;
  for (int i = threadIdx.x; i < nblk; i += blockDim.x) m = fmaxf(m, partial[i]);
  m = wave_max32(m);
  int wid = threadIdx.x >> 5;
  if ((threadIdx.x & 31) == 0) sm[wid] = m;
  __syncthreads();
  if (wid == 0) {
    float x = (threadIdx.x < (blockDim.x >> 5)) ? sm[threadIdx.x] : -INFINITY;
    x = wave_max32(x);
    if (threadIdx.x == 0) partial[nblk] = x;  // final max (before +1e-6)
  }
}

// ---------------- kernel 5: scale features once (matches reference rounding) ----

__global__ void __launch_bounds__(256) scale_kernel(const float* __restrict__ f,
                                                    const float* __restrict__ fmax_p,
                                                    float* __restrict__ fs, int total) {
  const float d = fmax_p[0] + 1e-6f;  // fmax = max(features) + 1e-6
  for (int i = blockIdx.x * blockDim.x + threadIdx.x; i < total; i += gridDim.x * blockDim.x)
    fs[i] = f[i] / d;
}

// ---------------- kernel 6: main score head (1 wave = 1 point, lane = channel) ----

// issue 4 row-half loads for one ID quad, directly in WMMA B layout:
// lane l -> channel l&15 (and +16) of row (l<16 ? even-slot : odd-slot)
#define ISSUE(IDS, LX, HX, LY, HY, VX, VY)                            \
  do {                                                                \
    int _sx = half ? (IDS).y : (IDS).x;                               \
    int _sy = half ? (IDS).w : (IDS).z;                               \
    VX = _sx >= 0; VY = _sy >= 0;                                     \
    int _ax = (_sx < 0 ? 0 : _sx) * Cn + cbase;                       \
    int _ay = (_sy < 0 ? 0 : _sy) * Cn + cbase;                       \
    LX = feats_s[_ax]; HX = feats_s[_ax + 16];                        \
    LY = feats_s[_ay]; HY = feats_s[_ay + 16];                        \
  } while (0)

// consume one staged quad into a dedicated accumulator pair (no reg rotation)
#define CONSUME(ALO, AHI, LX, HX, LY, HY, VX, VY)                     \
  do {                                                                \
    v2f _blo, _bhi;                                                   \
    _blo.x = VX ? LX : 0.0f; _bhi.x = VX ? HX : 0.0f;                 \
    _blo.y = VY ? LY : 0.0f; _bhi.y = VY ? HY : 0.0f;                 \
    mLo = fmaxf(mLo, VX ? LX : -INFINITY);                            \
    mLo = fmaxf(mLo, VY ? LY : -INFINITY);                            \
    mHi = fmaxf(mHi, VX ? HX : -INFINITY);                            \
    mHi = fmaxf(mHi, VY ? HY : -INFINITY);                            \
    ALO = __builtin_amdgcn_wmma_f32_16x16x4_f32(false, aones, false, _blo, \
                                                (short)0, ALO, false, false); \
    AHI = __builtin_amdgcn_wmma_f32_16x16x4_f32(false, aones, false, _bhi, \
                                                (short)0, AHI, false, false); \
  } while (0)

__global__ void __launch_bounds__(256) score_kernel(const float* __restrict__ feats_s,
                                                    const long long* __restrict__ idxs,
                                                    const int* __restrict__ hard,
                                                    const int* __restrict__ grid,
                                                    float* __restrict__ out) {
  __shared__ __align__(16) int lists[8][128];  // per-wave compacted neighbor ids (32 quads)

  const int lane = threadIdx.x & 31;
  const int wid = threadIdx.x >> 5;
  const int n = blockIdx.x * 8 + wid;  // grid = N/8 blocks, always in range

  const int b = (int)idxs[n * 3 + 0];
  const int h = (int)idxs[n * 3 + 1];
  const int w = (int)idxs[n * 3 + 2];
  const int base = b * HWn;

  // this point's scaled feature row, one channel per lane
  const float fown = feats_s[n * Cn + lane];

  // in-bounds window cell count (reduce_window pads with the init value)
  const int rlo = max(h - PAD, 0), rhi = min(h + PAD, Hn - 1);
  const int clo = max(w - PAD, 0), chi = min(w + PAD, Wn - 1);
  const int inb = (rhi - rlo + 1) * (chi - clo + 1);

  // init per-wave id list to -1 (pad sentinel); private to wave, DS ops in-order
  {
    int4 m1; m1.x = -1; m1.y = -1; m1.z = -1; m1.w = -1;
    ((int4*)(&lists[wid][0]))[lane] = m1;
  }

  // probe the 121 window cells: issue 4 masked loads first, then compact
  int idv[4];
  #pragma unroll
  for (int pass = 0; pass < 4; ++pass) {
    int j = pass * 32 + lane;
    int row = j / 11;
    int col = j - row * 11;
    int hh = h + row - PAD;
    int ww = w + col - PAD;
    bool valid = (j < 121) && ((unsigned)hh < (unsigned)Hn) && ((unsigned)ww < (unsigned)Wn);
    idv[pass] = valid ? grid[base + hh * Wn + ww] : -1;
  }
  int occ = 0;
  #pragma unroll
  for (int pass = 0; pass < 4; ++pass) {
    unsigned mask = (unsigned)__builtin_amdgcn_ballot_w32(idv[pass] >= 0);
    mask = (unsigned)__builtin_amdgcn_readfirstlane((int)mask);
    int pos = __popc(mask & ((1u << lane) - 1u));
    if (idv[pass] >= 0) lists[wid][occ + pos] = idv[pass];
    occ += __popc(mask);
  }

  // group loop: 2 quads (8 rows) per iteration; dedicated accs per stage
  const int ng = (occ + 3) >> 2;        // quads actually populated (>= 1)
  const int ngp = (ng + 1) & ~1;        // rounded up to even; pad quads are all -1
  const bool half = lane >= 16;
  const int cbase = lane & 15;

  v8f accLoA = {}, accHiA = {}, accLoB = {}, accHiB = {};
  const v2f aones = {1.0f, 1.0f};       // A (16x4) all ones -> D[m,c] = sum_k B[k,c]
  float mLo = -INFINITY, mHi = -INFINITY;

  float alx, ahx, aly, ahy; bool avx, avy;
  float blx, bhx, bly, bhy; bool bvx, bvy;

  int4 ia = *((const int4*)(&lists[wid][0]) + 0);
  int4 ib = *((const int4*)(&lists[wid][0]) + 1);
  ISSUE(ia, alx, ahx, aly, ahy, avx, avy);
  ISSUE(ib, blx, bhx, bly, bhy, bvx, bvy);

  for (int g = 0; g < ngp; g += 2) {    // wave-uniform: EXEC all-ones at every WMMA
    const bool more = (g + 2) < ngp;
    int4 na, nb;
    if (more) {
      na = *((const int4*)(&lists[wid][0]) + (g + 2));
      nb = *((const int4*)(&lists[wid][0]) + (g + 3));
    }
    CONSUME(accLoA, accHiA, alx, ahx, aly, ahy, avx, avy);
    if (more) ISSUE(na, alx, ahx, aly, ahy, avx, avy);
    CONSUME(accLoB, accHiB, blx, bhx, bly, bhy, bvx, bvy);
    if (more) ISSUE(nb, blx, bhx, bly, bhy, bvx, bvy);
  }

  // channel window-sum: lane c<16 reads accLo*[0] (M=0,N=c);
  // lane c>=16 reads accHi*[0] (duplicate M=8 row, N=c-16) => channel c
  float sum = (lane < 16) ? (accLoA[0] + accLoB[0]) : (accHiA[0] + accHiB[0]);

  // fold split-layout channel maxes: mLo covers slots{0,2}/{1,3} per half-wave
  float oLo = fmaxf(mLo, __shfl_xor(mLo, 16, 32));
  float oHi = fmaxf(mHi, __shfl_xor(mHi, 16, 32));
  float maxv = (lane < 16) ? oLo : oHi;
  if (occ < inb) maxv = fmaxf(maxv, 0.0f);  // empty in-bounds cells contribute 0.0

  const float valid_num = (float)occ + 1e-6f;
  const float mean = sum / valid_num;

  // softplus(dense - local_mean) at the point == jnp.logaddexp(x, 0)
  const float x = fown - mean;
  const float lms = fmaxf(x, 0.0f) + log1pf(expf(-fabsf(x)));

  // depth-wise max over channels (cross-lane)
  const float dwm = wave_max32(fown);
  const float sc = lms * (fown / (1e-6f + dwm));
  const float scores = wave_max32(sc);

  // local-max detection: any channel where own value equals window max
  unsigned det = (unsigned)__builtin_amdgcn_ballot_w32(fown == maxv);
  float result = det ? scores : 0.0f;
  if (hard[0] == 0) result = scores;

  if (lane == 0) out[n] = result;
}

// ---------------- launcher ----------------

extern "C" void kernel_launch(void* const* d_in, const int* in_sizes, int n_in,
                              void* d_out, int out_size, void* d_ws, size_t ws_size,
                              hipStream_t stream) {
  const float* features = (const float*)d_in[0];      // (N, C) f32
  const long long* idxs = (const long long*)d_in[1];  // (N, 3) i64
  const int* hard = (const int*)d_in[2];              // scalar
  float* out = (float*)d_out;                         // (N, 1) f32

  // workspace layout: [id grid: B*H*W ints][scaled feats: N*C floats][partials: 513 floats]
  int* grid = (int*)d_ws;
  float* feats_s = (float*)d_ws + (size_t)(Bn * HWn);
  float* partial = feats_s + (size_t)(Nn * Cn);

  init_grid_kernel<<<(Bn * HWn + 255) / 256, 256, 0, stream>>>(grid);
  scatter_ids_kernel<<<(Nn + 255) / 256, 256, 0, stream>>>(idxs, grid);
  reduce_max1_kernel<<<512, 256, 0, stream>>>(features, partial, Nn * Cn);
  reduce_max2_kernel<<<1, 256, 0, stream>>>(partial, 512);
  scale_kernel<<<512, 256, 0, stream>>>(features, partial + 512, feats_s, Nn * Cn);
  score_kernel<<<Nn / 8, 256, 0, stream>>>(feats_s, idxs, hard, grid, out);
}